// DiscoverODE_90761248899126
// MI455X (gfx1250) — compile-verified
//
#include <hip/hip_runtime.h>
#include <stdint.h>

// ---------------------------------------------------------------------------
// Coupled-pendulum ring ODE, fixed-step RK4 in float64.
//   dθ/dt = v
//   dv/dt = -ω0² sin θ + c (θ[n-1] - θ) + c (θ[n+1] - θ)
// Mapping: 1 wave32 = 1 batch row (512 pendula, 16 per lane, ring closed over
// the 32 lanes). Neighbor exchange via ds_bpermute (__shfl) -> no barriers in
// the 2048-step time loop. Input staged HBM->LDS via the gfx1250 Tensor Data
// Mover (tensor_load_to_lds + s_wait_tensorcnt).
// ---------------------------------------------------------------------------

#ifndef NSTEPS
#define NSTEPS 2048            // h = 2.0 / 2048 = 2^-10, exactly representable
#endif

#define NPEND   512            // pendula per row
#define CHUNK   16             // pendula per lane (32 lanes * 16 = 512)
#define ROWS_PB 8              // rows (waves) per 256-thread block

typedef __attribute__((ext_vector_type(4))) unsigned int v4u;
typedef __attribute__((ext_vector_type(4))) int          v4i;
typedef __attribute__((ext_vector_type(8))) int          v8i;

__device__ __forceinline__ void deriv16(const double (&sth)[CHUNK],
                                        double (&kv)[CHUNK],
                                        int lane, double w2, double cpl) {
    // Edge thetas from the ring neighbors (lane-1's element 15, lane+1's element 0).
    double tl = __shfl(sth[CHUNK - 1], (lane + 31) & 31, 32);
    double tr = __shfl(sth[0],         (lane + 1)  & 31, 32);
#pragma unroll
    for (int i = 0; i < CHUNK; ++i) {
        double lm = (i == 0)         ? tl : sth[i - 1];
        double rp = (i == CHUNK - 1) ? tr : sth[i + 1];
        kv[i] = -w2 * sin(sth[i]) + cpl * (lm + rp - 2.0 * sth[i]);
    }
}

__global__ __launch_bounds__(256)
void DiscoverODE_pendula_rk4(const float* __restrict__ x,
                             const double* __restrict__ omega0p,
                             const double* __restrict__ couplingp,
                             float* __restrict__ out) {
    __shared__ float sBuf[ROWS_PB * NPEND];          // 16 KB input tile

    const int lane = threadIdx.x & 31;
    const int wav  = threadIdx.x >> 5;
    const int row  = blockIdx.x * ROWS_PB + wav;     // batch row, 0..1023

    // ---- Stage the block's input tile into LDS with the Tensor Data Mover ----
    if (wav == 0) {                                  // wave-uniform branch
        uint64_t ga  = (uint64_t)(const void*)(x + (size_t)blockIdx.x * (ROWS_PB * NPEND));
        uint32_t lds = (uint32_t)(uintptr_t)(void*)&sBuf[0];

        // D# group 0: count=1, lds_addr, global_addr[56:0], type=2 ("image")
        v4u g0 = { 1u,
                   lds,
                   (uint32_t)ga,
                   (uint32_t)(((ga >> 32) & 0x01FFFFFFu) | 0x80000000u) };
        // D# group 1: data_size=4B, tensor_dim0=4096, tensor_dim1=1,
        //             tile_dim0=4096, tile_dim1=1, tensor_dim0_stride=4096
        v8i g1 = { (int)(2u << 16),                  // data_size = 2 -> 4 bytes
                   (int)((4096u & 0xFFFFu) << 16),   // tensor_dim0[15:0] in bits 63:48
                   (int)((4096u >> 16) | (1u << 16)),// tensor_dim0 hi | tensor_dim1 lo
                   (int)(4096u << 16),               // tile_dim0 in bits 127:112
                   (int)1,                           // tile_dim1 = 1
                   (int)4096,                        // tensor_dim0_stride lo32
                   0, 0 };
        v4i g2 = { 0, 0, 0, 0 };
        v4i g3 = { 0, 0, 0, 0 };
#if defined(__clang_major__) && (__clang_major__ >= 23)
        v8i g4 = { 0, 0, 0, 0, 0, 0, 0, 0 };
        __builtin_amdgcn_tensor_load_to_lds(g0, g1, g2, g3, g4, 0);
#else
        __builtin_amdgcn_tensor_load_to_lds(g0, g1, g2, g3, 0);
#endif
        __builtin_amdgcn_s_wait_tensorcnt(0);
    }
    __syncthreads();

    const double w0  = omega0p[0];
    const double cpl = couplingp[0];
    const double w2  = w0 * w0;

    constexpr double PI = 3.14159265358979323846;
    constexpr double h  = 2.0 / (double)NSTEPS;
    constexpr double h2 = 0.5 * h;
    constexpr double h6 = h / 6.0;

    // ---- Load this lane's 16 pendula from LDS, map [0,1] -> [-pi, pi] ----
    double th[CHUNK], v[CHUNK];
    {
        const float4* p4 = (const float4*)&sBuf[wav * NPEND + lane * CHUNK];
#pragma unroll
        for (int j = 0; j < CHUNK / 4; ++j) {
            float4 a = p4[j];
            th[4 * j + 0] = (double)a.x * (2.0 * PI) - PI;
            th[4 * j + 1] = (double)a.y * (2.0 * PI) - PI;
            th[4 * j + 2] = (double)a.z * (2.0 * PI) - PI;
            th[4 * j + 3] = (double)a.w * (2.0 * PI) - PI;
        }
#pragma unroll
        for (int i = 0; i < CHUNK; ++i) v[i] = 0.0;
    }

    // ---- Classic RK4, everything resident in VGPRs; shuffles only ----
    double sth[CHUNK], sv[CHUNK], ath[CHUNK], av[CHUNK], kv[CHUNK];
    for (int s = 0; s < NSTEPS; ++s) {
        // stage 1 at (th, v):   k1 = (v, f(th))
        deriv16(th, kv, lane, w2, cpl);
#pragma unroll
        for (int i = 0; i < CHUNK; ++i) {
            ath[i] = v[i];
            av[i]  = kv[i];
            sth[i] = th[i] + h2 * v[i];
            sv[i]  = v[i]  + h2 * kv[i];
        }
        // stage 2:  k2 = (sv, f(sth))
        deriv16(sth, kv, lane, w2, cpl);
#pragma unroll
        for (int i = 0; i < CHUNK; ++i) {
            ath[i] += 2.0 * sv[i];
            av[i]  += 2.0 * kv[i];
            double nt = th[i] + h2 * sv[i];
            double nv = v[i]  + h2 * kv[i];
            sth[i] = nt; sv[i] = nv;
        }
        // stage 3:  k3 = (sv, f(sth))
        deriv16(sth, kv, lane, w2, cpl);
#pragma unroll
        for (int i = 0; i < CHUNK; ++i) {
            ath[i] += 2.0 * sv[i];
            av[i]  += 2.0 * kv[i];
            double nt = th[i] + h * sv[i];
            double nv = v[i]  + h * kv[i];
            sth[i] = nt; sv[i] = nv;
        }
        // stage 4:  k4 = (sv, f(sth)); combine
        deriv16(sth, kv, lane, w2, cpl);
#pragma unroll
        for (int i = 0; i < CHUNK; ++i) {
            ath[i] += sv[i];
            av[i]  += kv[i];
            th[i]  += h6 * ath[i];
            v[i]   += h6 * av[i];
        }
    }

    // ---- Store final angles (output normalization is identity), f32 ----
    float4* o4 = (float4*)(out + (size_t)row * NPEND + lane * CHUNK);
#pragma unroll
    for (int j = 0; j < CHUNK / 4; ++j) {
        float4 a;
        a.x = (float)th[4 * j + 0];
        a.y = (float)th[4 * j + 1];
        a.z = (float)th[4 * j + 2];
        a.w = (float)th[4 * j + 3];
        o4[j] = a;
    }
}

extern "C" void kernel_launch(void* const* d_in, const int* in_sizes, int n_in,
                              void* d_out, int out_size, void* d_ws, size_t ws_size,
                              hipStream_t stream) {
    const float*  x        = (const float*)d_in[0];   // [1024, 512] f32
    const double* omega0   = (const double*)d_in[1];  // scalar f64
    const double* coupling = (const double*)d_in[2];  // scalar f64
    float*        out      = (float*)d_out;           // [1024, 512] f32

    const int rows   = in_sizes[0] / NPEND;           // 1024
    const int blocks = rows / ROWS_PB;                // 128
    DiscoverODE_pendula_rk4<<<blocks, 256, 0, stream>>>(x, omega0, coupling, out);
}